// BSI_71597104824914
// MI455X (gfx1250) — compile-verified
//
#include <hip/hip_runtime.h>
#include <hip/hip_bf16.h>
#include <cstdint>

// Problem constants (B, N, T, L, H) = (8, 128, 256, 32, 128)
#define BB   8
#define NN   128
#define TT   256
#define LL   32
#define HH   128
#define TTAU 224           // T - L
#define MT   14            // t-tiles of 16 over TTAU
#define HT   8             // h-tiles of 16 over H
#define KTOT (NN*HH)       // 16384, K dim of the G-GEMM
#define GSZ  (BB*TTAU*NN)  // 229376 elements of G (and of X_hat)

typedef __attribute__((ext_vector_type(2))) float v2f;
typedef __attribute__((ext_vector_type(8))) float v8f;
typedef __attribute__((ext_vector_type(4))) int   v4i;

// V_WMMA_F32_16X16X4_F32 : D(16x16 f32) = A(16x4 f32) x B(4x16 f32) + C
__device__ __forceinline__ v8f wmma4(v2f a, v2f b, v8f c) {
  return __builtin_amdgcn_wmma_f32_16x16x4_f32(
      /*neg_a=*/false, a, /*neg_b=*/false, b,
      /*c_mod=*/(short)0, c, /*reuse_a=*/false, /*reuse_b=*/false);
}

__device__ __forceinline__ float sigm(float v)  { return 1.0f / (1.0f + __expf(-v)); }
__device__ __forceinline__ float ftanh(float v) { return 1.0f - 2.0f / (__expf(2.0f * v) + 1.0f); }

// ---- gfx1250 async global->LDS copy (ASYNCcnt-tracked), with sync fallback --
#if defined(__has_builtin)
#if __has_builtin(__builtin_amdgcn_global_load_async_to_lds_b128) && \
    __has_builtin(__builtin_amdgcn_s_wait_asynccnt)
#define USE_ASYNC_LDS 1
#endif
#endif

#ifdef USE_ASYNC_LDS
typedef __attribute__((address_space(1))) v4i g_v4i;   // global (prints __device__)
typedef __attribute__((address_space(3))) v4i l_v4i;   // LDS
#endif

__device__ __forceinline__ void cp16(const float* __restrict__ g, float* l) {
#ifdef USE_ASYNC_LDS
  // flat LDS address low 32 bits == LDS byte offset (ISA aperture rule)
  __builtin_amdgcn_global_load_async_to_lds_b128(
      (g_v4i*)(uintptr_t)g, (l_v4i*)(uintptr_t)l,
      /*offset=*/0, /*cpol=*/0);
#else
  *(float4*)l = *(const float4*)g;
#endif
}

__device__ __forceinline__ void cp_wait_all() {
#ifdef USE_ASYNC_LDS
  __builtin_amdgcn_s_wait_asynccnt(0);
#endif
}

// ---------------------------------------------------------------------------
// Kernel 0a: Wsum[n,h] = sum_j W_fc[n,j,h];  bsum[n] = sum_j b_fc[n,j]
// ---------------------------------------------------------------------------
__global__ __launch_bounds__(256) void reduce_wfc_kernel(
    const float* __restrict__ W_fc, const float* __restrict__ b_fc,
    float* __restrict__ Wsum, float* __restrict__ bsum) {
  int gid = blockIdx.x * blockDim.x + threadIdx.x;   // 0 .. 16383
  int n = gid >> 7, h = gid & 127;
  const float* src = W_fc + (size_t)n * NN * HH + h;
  float acc = 0.0f;
  #pragma unroll 4
  for (int j = 0; j < NN; ++j) acc += src[(size_t)j * HH];
  Wsum[gid] = acc;
  if (gid < NN) {
    const float* br = b_fc + (size_t)gid * NN;
    float ba = 0.0f;
    #pragma unroll 4
    for (int j = 0; j < NN; ++j) ba += br[j];
    bsum[gid] = ba;
  }
}

// ---------------------------------------------------------------------------
// Kernel 0b: Bt[(n*H + h)*N + j] = W_fc[n, j, h]   (B matrix for the G-GEMM)
// ---------------------------------------------------------------------------
__global__ __launch_bounds__(256) void transpose_wfc_kernel(
    const float* __restrict__ W_fc, float* __restrict__ Bt) {
  int gid = blockIdx.x * blockDim.x + threadIdx.x;   // 0 .. N*H*N-1
  int j = gid & 127;
  int h = (gid >> 7) & 127;
  int n = gid >> 14;
  Bt[gid] = W_fc[((size_t)(n * NN + j)) * HH + h];
}

// ---------------------------------------------------------------------------
// Kernel 1: gates GEMM (WMMA f32, K=32) + LSTM activations.
//   one wave per (b, n, t-tile, h-tile); computes gi/gg/go 16x16 tiles
//   (gf is unused by the reference and skipped entirely);
//   stores hs[b,t, n*H + h] = h / x[b,n,L+t]  (A matrix of the G-GEMM).
// ---------------------------------------------------------------------------
__global__ __launch_bounds__(256) void lstm_gates_kernel(
    const float* __restrict__ x, const float* __restrict__ W_ih,
    const float* __restrict__ b_ih, const float* __restrict__ b_hh,
    float* __restrict__ hs) {
  int w    = (blockIdx.x * blockDim.x + threadIdx.x) >> 5;
  int lane = threadIdx.x & 31;
  int ht = w & 7;  w >>= 3;
  int mt = w % MT; w /= MT;
  int n  = w & 127; w >>= 7;
  int b  = w;
  if (b >= BB) return;

  const int t0 = mt * 16, h0 = ht * 16;
  const int Ml = lane & 15;            // A row (t) / B-C-D column (h)
  const int kb = (lane >> 4) << 1;     // K sub-offset per ISA A/B layout

  const float* xbn = x    + (size_t)(b * NN + n) * TT;
  const float* wih = W_ih + (size_t)n * (4 * HH) * LL;

  v8f ci = {}, cg = {}, co = {};
  #pragma unroll
  for (int kk = 0; kk < 8; ++kk) {
    const int k = kk * 4 + kb;
    v2f a;
    a.x = xbn[t0 + Ml + k];
    a.y = xbn[t0 + Ml + k + 1];
    const float* wg = wih + (size_t)(h0 + Ml) * LL + k;  // g = h0+Ml (gi)
    v2f bi = *(const v2f*)(wg);                          //   l = k, k+1
    v2f bg = *(const v2f*)(wg + (size_t)2 * HH * LL);    // g += 2H (gg)
    v2f bo = *(const v2f*)(wg + (size_t)3 * HH * LL);    // g += 3H (go)
    ci = wmma4(a, bi, ci);
    cg = wmma4(a, bg, cg);
    co = wmma4(a, bo, co);
  }

  const float bia = b_ih[n * 512 + h0 + Ml]       + b_hh[n * 512 + h0 + Ml];
  const float big = b_ih[n * 512 + 256 + h0 + Ml] + b_hh[n * 512 + 256 + h0 + Ml];
  const float bio = b_ih[n * 512 + 384 + h0 + Ml] + b_hh[n * 512 + 384 + h0 + Ml];

  const int rowadd = (lane >> 4) << 3;  // lanes 16-31 hold rows M+8
  #pragma unroll
  for (int r = 0; r < 8; ++r) {
    const int t  = t0 + r + rowadd;
    const float gi = ci[r] + bia;
    const float gg = cg[r] + big;
    const float go = co[r] + bio;
    const float cc = sigm(gi) * ftanh(gg);
    const float hv = sigm(go) * ftanh(cc);
    const float hsv = hv / xbn[LL + t];
    hs[((size_t)(b * TTAU + t) * NN + n) * HH + h0 + Ml] = hsv;
  }
}

// ---------------------------------------------------------------------------
// Kernel 2: G[b,t,j] = (1/N) * [ hs(224 x 16384) @ Bt(16384 x 128)
//                               + invx(224 x 128) @ b_fc(128 x 128) ]
//   4 waves / block; block owns a 16(t) x 64(j) tile.  A (16x64) and B (64x64)
//   K-chunks are double-buffered in LDS via GLOBAL_LOAD_ASYNC_TO_LDS_B128,
//   overlapping chunk k+1's DMA with chunk k's 16 WMMA K-steps.
// ---------------------------------------------------------------------------
__global__ __launch_bounds__(128) void g_gemm_kernel(
    const float* __restrict__ hs, const float* __restrict__ Bt,
    const float* __restrict__ x, const float* __restrict__ b_fc,
    float* __restrict__ G) {
  __shared__ float As[2][16][68];  // 16 t-rows x 64-K chunk (stride 68: conflict-free v2f)
  __shared__ float Bs[2][64][80];  // 64 K-rows x 64 j     (stride 80: half-waves hit
                                   //  disjoint bank ranges: 2*80 % 64 == 32)

  int bi = blockIdx.x;
  const int jg = bi & 1;        bi >>= 1;
  const int mt = bi % MT;
  const int b  = bi / MT;
  const int tid  = threadIdx.x;
  const int wave = tid >> 5;
  const int lane = tid & 31;

  const int t0   = mt * 16;
  const int jb0  = jg * 64;          // block's j base
  const int jloc = wave * 16;        // wave's j-tile within the block
  const int Ml   = lane & 15;
  const int kb   = (lane >> 4) << 1;

  // cooperative loader assignment
  const int arow = tid >> 3;           // A: 16 rows x 2 float4 per thread
  const int ac0  = (tid & 7) * 8;
  const float* agbase = hs + (size_t)(b * TTAU + t0 + arow) * KTOT;

  v8f acc = {};

  auto issue_chunk = [&](int kc, int buf) {
    cp16(agbase + kc + ac0,     &As[buf][arow][ac0]);
    cp16(agbase + kc + ac0 + 4, &As[buf][arow][ac0 + 4]);
    #pragma unroll
    for (int i = 0; i < 8; ++i) {      // B: 1024 float4 segs, 8 per thread
      const int s    = i * 128 + tid;  // lane-consecutive segs -> coalesced
      const int brow = s >> 4;
      const int bcol = (s & 15) * 4;
      cp16(Bt + (size_t)(kc + brow) * NN + jb0 + bcol, &Bs[buf][brow][bcol]);
    }
  };

  issue_chunk(0, 0);
  int buf = 0;
  for (int kc = 0; kc < KTOT; kc += 64) {
    cp_wait_all();       // each wave's async writes complete ...
    __syncthreads();     // ... and are made visible block-wide
    if (kc + 64 < KTOT) issue_chunk(kc + 64, buf ^ 1);
    #pragma unroll
    for (int kk = 0; kk < 16; ++kk) {
      const int k = kk * 4 + kb;
      v2f a = *(const v2f*)&As[buf][Ml][k];
      v2f bf;
      bf.x = Bs[buf][k][jloc + Ml];
      bf.y = Bs[buf][k + 1][jloc + Ml];
      acc = wmma4(a, bf, acc);
    }
    buf ^= 1;
  }

  // bias GEMM:  Σ_n b_fc[n,j] / x[b,n,L+t]   (K = 128)
  #pragma unroll 4
  for (int kk = 0; kk < 32; ++kk) {
    const int k = kk * 4 + kb;
    v2f a;
    a.x = 1.0f / x[(size_t)(b * NN + k) * TT + LL + t0 + Ml];
    a.y = 1.0f / x[(size_t)(b * NN + k + 1) * TT + LL + t0 + Ml];
    v2f bf;
    bf.x = b_fc[(size_t)k * NN + jb0 + jloc + Ml];
    bf.y = b_fc[(size_t)(k + 1) * NN + jb0 + jloc + Ml];
    acc = wmma4(a, bf, acc);
  }

  const int rowadd = (lane >> 4) << 3;
  #pragma unroll
  for (int r = 0; r < 8; ++r) {
    const int t = t0 + r + rowadd;
    G[(size_t)(b * TTAU + t) * NN + jb0 + jloc + Ml] = acc[r] * (1.0f / (float)NN);
  }
}

// ---------------------------------------------------------------------------
// Kernel 3: X_hat[b,t,n] = x[b,n,L+t] * <hs[b,t,n,:], Wsum[n,:]> + bsum[n] + 1e-6
// ---------------------------------------------------------------------------
__global__ __launch_bounds__(256) void xhat_kernel(
    const float* __restrict__ hs, const float* __restrict__ x,
    const float* __restrict__ Wsum, const float* __restrict__ bsum,
    float* __restrict__ Xh) {
  int gid = blockIdx.x * blockDim.x + threadIdx.x;   // 0 .. GSZ-1
  const int n = gid & 127;
  const int t = (gid >> 7) % TTAU;
  const int b = gid / (NN * TTAU);
  const float* hr = hs + (size_t)gid * HH;
  const float* wr = Wsum + (size_t)n * HH;
  float acc = 0.0f;
  #pragma unroll
  for (int h = 0; h < HH; h += 4) {
    float4 hv = *(const float4*)(hr + h);
    float4 wv = *(const float4*)(wr + h);
    acc = fmaf(hv.x, wv.x, acc);
    acc = fmaf(hv.y, wv.y, acc);
    acc = fmaf(hv.z, wv.z, acc);
    acc = fmaf(hv.w, wv.w, acc);
  }
  const float xv = x[(size_t)(b * NN + n) * TT + LL + t];
  Xh[gid] = xv * acc + bsum[n] + 1e-6f;
}

// ---------------------------------------------------------------------------
extern "C" void kernel_launch(void* const* d_in, const int* in_sizes, int n_in,
                              void* d_out, int out_size, void* d_ws, size_t ws_size,
                              hipStream_t stream) {
  const float* x    = (const float*)d_in[0];
  const float* W_ih = (const float*)d_in[1];
  const float* b_ih = (const float*)d_in[2];
  const float* b_hh = (const float*)d_in[3];
  const float* W_fc = (const float*)d_in[4];
  const float* b_fc = (const float*)d_in[5];
  float* out = (float*)d_out;

  // workspace layout (floats): hs | Bt | Wsum | bsum  (~126 MB total)
  float* hs   = (float*)d_ws;                      // B*TTAU*N*H = 29,360,128
  float* Bt   = hs + (size_t)BB * TTAU * NN * HH;  // N*H*N      =  2,097,152
  float* Wsum = Bt + (size_t)NN * HH * NN;         // N*H        =     16,384
  float* bsum = Wsum + NN * HH;                    // N          =        128

  reduce_wfc_kernel<<<(NN * HH) / 256, 256, 0, stream>>>(W_fc, b_fc, Wsum, bsum);
  transpose_wfc_kernel<<<(NN * HH * NN) / 256, 256, 0, stream>>>(W_fc, Bt);
  // 8*128*14*8 = 114,688 wave jobs, 8 waves per 256-thread block
  lstm_gates_kernel<<<(BB * NN * MT * HT) / 8, 256, 0, stream>>>(x, W_ih, b_ih, b_hh, hs);
  // 8 batches * 14 m-tiles * 2 j-groups = 224 blocks of 4 waves
  g_gemm_kernel<<<BB * MT * 2, 128, 0, stream>>>(hs, Bt, x, b_fc, out);
  xhat_kernel<<<GSZ / 256, 256, 0, stream>>>(hs, x, Wsum, bsum, out + GSZ);
}